// InvertedDotProductAttention_26817775796317
// MI455X (gfx1250) — compile-verified
//
#include <hip/hip_runtime.h>
#include <math.h>

typedef __attribute__((ext_vector_type(2))) float v2f;
typedef __attribute__((ext_vector_type(4))) float v4f;
typedef __attribute__((ext_vector_type(8))) float v8f;

#define BATCH 64
#define QN 64
#define DH 64
#define KTOT 8192
#define KT 128
#define PAD 68      // row stride for Qs/Ks/Vs: 4-bank skew -> conflict-free frag loads
#define SPAD 132    // row stride for S tile
#define EPSV 1e-8f

// ---------------------------------------------------------------------------
// Main fused kernel: one (batch, 128-key tile) per 256-thread block (8 waves).
//  K/V tiles staged with GLOBAL_LOAD_ASYNC_TO_LDS_B128 (ASYNCcnt),
//  S = (Q/8) @ K^T via v_wmma_f32_16x16x4_f32, column softmax over q in LDS,
//  unnormalized attn -> global, partial rowsum + partial O=attn@V -> atomics.
// ---------------------------------------------------------------------------
__global__ __launch_bounds__(256) void ida_main(
    const float* __restrict__ qg, const float* __restrict__ kg,
    const float* __restrict__ vg, float* __restrict__ attn_out,
    float* __restrict__ rowsum, float* __restrict__ oacc)
{
    __shared__ float Qs[QN][PAD];
    __shared__ float Ks[KT][PAD];
    __shared__ float Vs[KT][PAD];
    __shared__ float S[QN][SPAD];

    const int t    = threadIdx.x;
    const int b    = blockIdx.y;
    const int k0   = blockIdx.x * KT;
    const int lane = t & 31;
    const int w    = t >> 5;            // wave id 0..7
    const int l16  = lane & 15;
    const int kid  = (lane >> 4) << 1;  // 0 or 2 (A/B frag K sub-index)
    const int hi8  = (lane >> 4) << 3;  // 0 or 8 (C frag M offset)

    // ---- async stage K/V tiles: global -> LDS directly (no VGPR staging) ----
    {
        const float* kb = kg + ((size_t)b * KTOT + k0) * DH;
        const float* vb = vg + ((size_t)b * KTOT + k0) * DH;
        for (int i = t; i < KT * DH / 4; i += 256) {
            int r = i >> 4, c = (i & 15) << 2;
            const float* ksrc = kb + (size_t)r * DH + c;
            const float* vsrc = vb + (size_t)r * DH + c;
            // LDS aperture: low 32 bits of generic pointer == wave LDS offset
            unsigned kdst = (unsigned)(size_t)&Ks[r][c];
            unsigned vdst = (unsigned)(size_t)&Vs[r][c];
            asm volatile("global_load_async_to_lds_b128 %0, %1, off"
                         :: "v"(kdst), "v"(ksrc) : "memory");
            asm volatile("global_load_async_to_lds_b128 %0, %1, off"
                         :: "v"(vdst), "v"(vsrc) : "memory");
        }
    }

    // ---- stage Q with 1/sqrt(d) scaling (small: 16 KB) ----
    {
        const float* qb = qg + (size_t)b * QN * DH;
        for (int i = t; i < QN * DH / 4; i += 256) {
            int r = i >> 4, c = (i & 15) << 2;
            v4f val = *(const v4f*)(qb + r * DH + c);
            val *= 0.125f;
            *(v4f*)&Qs[r][c] = val;
        }
    }

    asm volatile("s_wait_asynccnt 0x0" ::: "memory");
    __syncthreads();

    // ---- GEMM1: S[64 x 128] = Qs @ Ks^T. wave w owns N-strip w. ----
    {
        v8f acc[4] = {};                    // 4 M-tiles (q = 0..63)
        const int nrow = w * 16 + l16;      // key row for B fragment
        for (int kk = 0; kk < DH; kk += 4) {
            v2f bf;
            bf.x = Ks[nrow][kk + kid];
            bf.y = Ks[nrow][kk + kid + 1];
#pragma unroll
            for (int mt = 0; mt < 4; ++mt) {
                v2f af;
                af.x = Qs[mt * 16 + l16][kk + kid];
                af.y = Qs[mt * 16 + l16][kk + kid + 1];
                acc[mt] = __builtin_amdgcn_wmma_f32_16x16x4_f32(
                    false, af, false, bf, (short)0, acc[mt], false, false);
            }
        }
#pragma unroll
        for (int mt = 0; mt < 4; ++mt)
#pragma unroll
            for (int r = 0; r < 8; ++r)
                S[mt * 16 + r + hi8][w * 16 + l16] = acc[mt][r];
    }
    __syncthreads();

    // ---- column softmax over q (64 values per key column), clip [0,1] ----
    if (t < KT) {
        float m = -3.4e38f;
        for (int qq = 0; qq < QN; ++qq) m = fmaxf(m, S[qq][t]);
        float s = 0.f;
        for (int qq = 0; qq < QN; ++qq) {
            float e = __expf(S[qq][t] - m);
            S[qq][t] = e; s += e;
        }
        float inv = 1.f / s;
        for (int qq = 0; qq < QN; ++qq) S[qq][t] = fminf(S[qq][t] * inv, 1.0f);
    }
    __syncthreads();

    // ---- partial row sums over this key tile -> global atomic ----
    if (t < QN) {
        float s = 0.f;
        for (int c = 0; c < KT; ++c) s += S[t][c];
        atomicAdd(&rowsum[b * QN + t], s);
    }

    // ---- stream unnormalized attn tile to global (rescaled later; keep RT
    //      so it stays L2-resident for the rescale pass) ----
    {
        float* ao = attn_out + (size_t)b * QN * KTOT + k0;
        for (int i = t; i < QN * KT / 4; i += 256) {
            int r = i >> 5, c = (i & 31) << 2;
            v4f val;
            val.x = S[r][c];     val.y = S[r][c + 1];
            val.z = S[r][c + 2]; val.w = S[r][c + 3];
            *(v4f*)(ao + (size_t)r * KTOT + c) = val;
        }
    }

    // ---- GEMM2: O_partial[64 x 64] = attn_tile @ V_tile. 2 tiles/wave ----
    {
        const int mt  = w >> 1;
        const int ntb = (w & 1) * 2;
        v8f acc[2] = {};
        for (int kk = 0; kk < KT; kk += 4) {
            v2f af;
            af.x = S[mt * 16 + l16][kk + kid];
            af.y = S[mt * 16 + l16][kk + kid + 1];
#pragma unroll
            for (int j = 0; j < 2; ++j) {
                v2f bf;
                bf.x = Vs[kk + kid][(ntb + j) * 16 + l16];
                bf.y = Vs[kk + kid + 1][(ntb + j) * 16 + l16];
                acc[j] = __builtin_amdgcn_wmma_f32_16x16x4_f32(
                    false, af, false, bf, (short)0, acc[j], false, false);
            }
        }
        float* ob = oacc + (size_t)b * QN * DH;
#pragma unroll
        for (int j = 0; j < 2; ++j)
#pragma unroll
            for (int r = 0; r < 8; ++r)
                atomicAdd(&ob[(mt * 16 + r + hi8) * DH + (ntb + j) * 16 + l16],
                          acc[j][r]);
    }
}

// ---- workspace zero-init (rowsum + O accumulator) ----
__global__ void ida_init(float* __restrict__ ws, int n) {
    int i = blockIdx.x * 256 + threadIdx.x;
    if (i < n) ws[i] = 0.f;
}

// ---- out[b,q,d] = O[b,q,d] / (rowsum[b,q] + eps) ----
__global__ void ida_out(const float* __restrict__ oacc,
                        const float* __restrict__ rowsum,
                        float* __restrict__ out) {
    int i = blockIdx.x * 256 + threadIdx.x;          // over B*QN*DH
    int bq = i >> 6;
    float sc = 1.f / (rowsum[bq] + EPSV);
    __builtin_nontemporal_store(oacc[i] * sc, &out[i]);
}

// ---- attn[b,q,k] /= (rowsum[b,q] + eps), vectorized float4 ----
__global__ void ida_scale(float* __restrict__ attn,
                          const float* __restrict__ rowsum) {
    size_t f = ((size_t)blockIdx.x * 256 + threadIdx.x) * 4;
    int bq = (int)(f >> 13);                         // f / 8192 = b*64 + q
    float sc = 1.f / (rowsum[bq] + EPSV);
    v4f v = *(const v4f*)(attn + f);                 // expect L2 hit
    v *= sc;
    __builtin_nontemporal_store(v, (v4f*)(attn + f));
}

extern "C" void kernel_launch(void* const* d_in, const int* in_sizes, int n_in,
                              void* d_out, int out_size, void* d_ws, size_t ws_size,
                              hipStream_t stream) {
    const float* q = (const float*)d_in[0];
    const float* k = (const float*)d_in[1];
    const float* v = (const float*)d_in[2];

    float* out    = (float*)d_out;                        // B*QN*DH
    float* attn   = out + (size_t)BATCH * QN * DH;        // B*QN*KTOT
    float* rowsum = (float*)d_ws;                         // B*QN
    float* oacc   = rowsum + BATCH * QN;                  // B*QN*DH

    int nws = BATCH * QN + BATCH * QN * DH;
    ida_init<<<(nws + 255) / 256, 256, 0, stream>>>(rowsum, nws);

    ida_main<<<dim3(KTOT / KT, BATCH), 256, 0, stream>>>(q, k, v, attn,
                                                         rowsum, oacc);

    ida_out<<<(BATCH * QN * DH) / 256, 256, 0, stream>>>(oacc, rowsum, out);

    ida_scale<<<(int)(((size_t)BATCH * QN * KTOT / 4) / 256), 256, 0, stream>>>(
        attn, rowsum);
}